// YOLOv1Loss_74088185856312
// MI455X (gfx1250) — compile-verified
//
#include <hip/hip_runtime.h>
#include <hip/hip_bf16.h>
#include <stdint.h>

// ---------------- problem constants ----------------
#define SQ          49          // S*S
#define NBATCH      16384
#define PC          30          // 5*B + C per cell (pred)
#define TC          25          // 5 + C per cell (target)
#define CPB         256         // cells per block (== blockDim.x)
#define NBLOCKS     (NBATCH * SQ / CPB)   // 3136, exact

typedef int v4i __attribute__((__vector_size__(4 * sizeof(int))));
typedef __attribute__((address_space(1))) v4i glb_v4i;
typedef __attribute__((address_space(3))) v4i lds_v4i;

// 16-byte global -> LDS async copy with compile-time immediate offset.
// ISA: INST_OFFSET is added to BOTH the global and the LDS address, so one
// (gaddr, ldsaddr) VGPR pair serves a whole batch of copies 4096 B apart.
template <int OFF>
__device__ __forceinline__ void async_copy_b128_off(const void* g, void* l) {
#if __has_builtin(__builtin_amdgcn_global_load_async_to_lds_b128)
  __builtin_amdgcn_global_load_async_to_lds_b128(
      (glb_v4i*)(unsigned long long)(g),
      (lds_v4i*)(unsigned int)(unsigned long long)(l), OFF, 0);
#else
  unsigned int lds_off = (unsigned int)(unsigned long long)(l);
  asm volatile("global_load_async_to_lds_b128 %0, %1, off offset:%c2"
               :: "v"(lds_off), "v"(g), "i"(OFF) : "memory");
#endif
}

__device__ __forceinline__ void wait_asynccnt0() {
#if __has_builtin(__builtin_amdgcn_s_wait_asynccnt)
  __builtin_amdgcn_s_wait_asynccnt(0);
#else
  asm volatile("s_wait_asynccnt 0" ::: "memory");
#endif
}

// ---------------- main loss kernel: one cell per thread ----------------
__global__ __launch_bounds__(CPB) void yolo_loss_main(
    const float* __restrict__ pred, const float* __restrict__ target,
    float* __restrict__ partial) {
  __shared__ float sp[CPB * PC];   // 30720 B = 1920 x b128
  __shared__ float st[CPB * TC];   // 25600 B = 1600 x b128
  __shared__ float wsum[8];

  const int tid = threadIdx.x;
  const int m0  = blockIdx.x * CPB;
  const float* gp = pred   + (size_t)m0 * PC;
  const float* gt = target + (size_t)m0 * TC;

  // ---- async global->LDS streaming: fixed per-thread base, imm offsets ----
  {
    const char* gb = (const char*)gp + tid * 16;
    char*       lb = (char*)sp + tid * 16;
    async_copy_b128_off<0     >(gb, lb);
    async_copy_b128_off<4096  >(gb, lb);
    async_copy_b128_off<8192  >(gb, lb);
    async_copy_b128_off<12288 >(gb, lb);
    async_copy_b128_off<16384 >(gb, lb);
    async_copy_b128_off<20480 >(gb, lb);
    async_copy_b128_off<24576 >(gb, lb);
    if (tid < 128) async_copy_b128_off<28672>(gb, lb);   // 30720 B total
  }
  {
    const char* gb = (const char*)gt + tid * 16;
    char*       lb = (char*)st + tid * 16;
    async_copy_b128_off<0     >(gb, lb);
    async_copy_b128_off<4096  >(gb, lb);
    async_copy_b128_off<8192  >(gb, lb);
    async_copy_b128_off<12288 >(gb, lb);
    async_copy_b128_off<16384 >(gb, lb);
    async_copy_b128_off<20480 >(gb, lb);
    if (tid < 64) async_copy_b128_off<24576>(gb, lb);    // 25600 B total
  }
  wait_asynccnt0();
  __syncthreads();

  const int m = m0 + tid;
  const float* P = &sp[tid * PC];  // stride 30 dwords: conflict-free in wave32
  const float* T = &st[tid * TC];  // stride 25 dwords: conflict-free in wave32

  const float conf_t = T[0];
  const float obj    = (conf_t == 1.0f) ? 1.0f : 0.0f;
  const float noobj  = (conf_t == 0.0f) ? 1.0f : 0.0f;

  const int grid = m % SQ;
  const float gx = (float)(grid % 7);
  const float gy = (float)(grid / 7);

  // target box
  const float tx = T[1], ty = T[2], tw = T[3], th = T[4];
  const float txc = (gx + tx) / 7.0f, tyc = (gy + ty) / 7.0f;
  const float tx1 = txc - tw * 0.5f, ty1 = tyc - th * 0.5f;
  const float tx2 = txc + tw * 0.5f, ty2 = tyc + th * 0.5f;
  const float area_t = (tx2 - tx1) * (ty2 - ty1);

  float pconf[2], px[2], py[2], pw[2], ph[2], iou[2];
#pragma unroll
  for (int b = 0; b < 2; ++b) {
    pconf[b] = fabsf(P[5 * b + 0]);
    px[b]    = fabsf(P[5 * b + 1]);
    py[b]    = fabsf(P[5 * b + 2]);
    pw[b]    = fabsf(P[5 * b + 3]);
    ph[b]    = fabsf(P[5 * b + 4]);
    const float xc = (gx + px[b]) / 7.0f, yc = (gy + py[b]) / 7.0f;
    const float x1 = xc - pw[b] * 0.5f, y1 = yc - ph[b] * 0.5f;
    const float x2 = xc + pw[b] * 0.5f, y2 = yc + ph[b] * 0.5f;
    const float ltx = fmaxf(x1, tx1), lty = fmaxf(y1, ty1);
    const float rbx = fminf(x2, tx2), rby = fminf(y2, ty2);
    const float iw = fmaxf(rbx - ltx, 0.0f);
    const float ih = fmaxf(rby - lty, 0.0f);
    const float inter  = iw * ih;
    const float area_p = (x2 - x1) * (y2 - y1);
    iou[b] = inter / (area_p + area_t - inter);
  }

  // argmax with first-max tie-break (matches jnp.argmax)
  const int   best    = (iou[1] > iou[0]) ? 1 : 0;
  const float max_iou = fmaxf(iou[0], iou[1]);

  const float rx = px[best], ry = py[best];
  const float rw = pw[best], rh = ph[best];
  const float rc = pconf[best];

  const float dx = rx - tx, dy = ry - ty;
  const float dw = sqrtf(rw) - sqrtf(tw);
  const float dh = sqrtf(rh) - sqrtf(th);
  const float loc = 5.0f * (dx * dx + dy * dy + dw * dw + dh * dh);

  const float dc = rc - max_iou;

  float cls = 0.0f;
#pragma unroll
  for (int c = 0; c < 20; ++c) {
    const float d = fabsf(P[10 + c]) - T[5 + c];
    cls += d * d;
  }

  float loss = obj * (loc + dc * dc + cls) +
               0.5f * noobj * (pconf[0] * pconf[0] + pconf[1] * pconf[1]);

  // wave32 reduction (deterministic), then fixed-order block reduction
#pragma unroll
  for (int off = 16; off > 0; off >>= 1)
    loss += __shfl_xor(loss, off, 32);
  if ((tid & 31) == 0) wsum[tid >> 5] = loss;
  __syncthreads();
  if (tid == 0) {
    float s = 0.0f;
#pragma unroll
    for (int w = 0; w < 8; ++w) s += wsum[w];
    partial[blockIdx.x] = s;
  }
}

// ---------------- deterministic final reduction ----------------
__global__ __launch_bounds__(256) void yolo_loss_reduce(
    const float* __restrict__ partial, float* __restrict__ out, int n) {
  __shared__ float sh[256];
  float s = 0.0f;
  for (int i = threadIdx.x; i < n; i += 256) s += partial[i];  // fixed order
  sh[threadIdx.x] = s;
  __syncthreads();
  for (int w = 128; w > 0; w >>= 1) {
    if ((int)threadIdx.x < w) sh[threadIdx.x] += sh[threadIdx.x + w];
    __syncthreads();
  }
  if (threadIdx.x == 0) out[0] = sh[0] * (1.0f / (float)NBATCH);
}

extern "C" void kernel_launch(void* const* d_in, const int* in_sizes, int n_in,
                              void* d_out, int out_size, void* d_ws, size_t ws_size,
                              hipStream_t stream) {
  const float* pred   = (const float*)d_in[0];
  const float* target = (const float*)d_in[1];
  float* out = (float*)d_out;
  float* ws  = (float*)d_ws;   // needs NBLOCKS floats = 12.25 KB

  yolo_loss_main<<<NBLOCKS, CPB, 0, stream>>>(pred, target, ws);
  yolo_loss_reduce<<<1, 256, 0, stream>>>(ws, out, NBLOCKS);
}